// FlowActionExpert_32796370272615
// MI455X (gfx1250) — compile-verified
//
#include <hip/hip_runtime.h>
#include <hip/hip_bf16.h>
#include <math.h>

// ---------------------------------------------------------------------------
// FlowActionExpert forward for MI455X (gfx1250), fp32 end-to-end.
// GEMMs: V_WMMA_F32_16X16X4_F32, double-buffered LDS tiles, both A and W
// tiles staged with GLOBAL_LOAD_ASYNC_TO_LDS_B64/B128 (ASYNCcnt) on the
// full-tile fast path; guarded scalar staging only on ragged edges.
// ---------------------------------------------------------------------------

#define Bsz 4
#define CHUNK 24
#define LTOK 26            // 2 prefix tokens + CHUNK
#define DM 768
#define NH 12
#define HD 64
#define DST 64             // D_STATE
#define DIN 1536           // D_INNER
#define DTR 48             // DT_RANK
#define XPN (DTR + 2*DST)  // 176
#define MLPH 3072

#define WS_STR 40          // W tile row stride (floats): 160B, 16B aligned rows
#define AS_STR 36          // A tile row stride (floats): 144B, 8B aligned rows

typedef float v2f __attribute__((ext_vector_type(2)));
typedef float v8f __attribute__((ext_vector_type(8)));

__device__ __forceinline__ float sigmoidf_(float x) { return 1.f / (1.f + expf(-x)); }

template<int ACT> __device__ __forceinline__ float apply_act(float x) {
  if constexpr (ACT == 1) return x * sigmoidf_(x);                      // silu
  else if constexpr (ACT == 2) return 0.5f * x * (1.f + erff(x * 0.70710678118f)); // gelu exact
  else if constexpr (ACT == 3) return (x > 20.f) ? x : log1pf(expf(x)); // softplus
  else return x;
}

// ---------------------------------------------------------------------------
// Generic GEMM: C[M,N] = act( A[M,K] * W[N,K]^T + bias ), optional C += result
// Block: 256 threads (8 waves). Tile: 16 (M) x 128 (N), K chunks of 32.
// Each wave computes one 16x16 tile with 8 x V_WMMA_F32_16X16X4_F32 per chunk.
// Double-buffered LDS; tiles staged via async global->LDS copies.
// ---------------------------------------------------------------------------
template<int ACT, bool ACC, bool HAS_BIAS>
__global__ void __launch_bounds__(256)
gemm_wmma(const float* __restrict__ A, int lda,
          const float* __restrict__ W,
          const float* __restrict__ bias,
          float* __restrict__ C, int ldc,
          int M, int N, int K)
{
  __shared__ __align__(16) float As[2][16][AS_STR];
  __shared__ __align__(16) float Ws[2][128][WS_STR];
  const int n0 = blockIdx.x * 128;
  const int m0 = blockIdx.y * 16;
  const int t = threadIdx.x;
  const int wave = t >> 5;
  const int lane = t & 31;
  const bool fullN = (n0 + 128 <= N);
  const bool fullM = (m0 + 16 <= M);

  // staging helpers -------------------------------------------------------
  auto stage_A = [&](int buf, int k0) {
    int m = t >> 4;            // 0..15
    int kk = (t & 15) * 2;     // 0..30
    if (fullM && (k0 + 32 <= K)) {
      // fast path: one async b64 global->LDS per thread (ASYNCcnt tracked)
      unsigned lds_addr = (unsigned)(uintptr_t)&As[buf][m][kk];
      unsigned goff = (unsigned)(((long)(m0 + m) * lda + k0 + kk) * 4);
      asm volatile("global_load_async_to_lds_b64 %0, %1, %2"
                   :: "v"(lds_addr), "v"(goff), "s"(A) : "memory");
    } else {
      int gm = m0 + m;
      float a0 = 0.f, a1 = 0.f;
      if (gm < M) {
        if (k0 + kk     < K) a0 = A[(long)gm * lda + k0 + kk];
        if (k0 + kk + 1 < K) a1 = A[(long)gm * lda + k0 + kk + 1];
      }
      As[buf][m][kk] = a0; As[buf][m][kk + 1] = a1;
    }
  };

  auto stage_W = [&](int buf, int k0) {
    int r = t >> 1;            // 0..127
    int half = t & 1;          // 16-float half of the row
    if (fullN && (k0 + 32 <= K)) {
      // fast path: 4x async b128 global->LDS per thread (ASYNCcnt tracked)
      unsigned lds_base = (unsigned)(uintptr_t)&Ws[buf][r][half * 16];
      long gelem = (long)(n0 + r) * K + k0 + half * 16;
      #pragma unroll
      for (int i = 0; i < 4; ++i) {
        unsigned lds_addr = lds_base + i * 16;
        unsigned goff = (unsigned)((gelem + i * 4) * 4);
        asm volatile("global_load_async_to_lds_b128 %0, %1, %2"
                     :: "v"(lds_addr), "v"(goff), "s"(W) : "memory");
      }
    } else {
      // ragged tiles: guarded scalar staging
      int halfk = half * 16;
      int gn = n0 + r;
      #pragma unroll
      for (int i = 0; i < 16; ++i) {
        int kk = halfk + i;
        float w = 0.f;
        if (gn < N && (k0 + kk) < K) w = W[(long)gn * K + k0 + kk];
        Ws[buf][r][kk] = w;
      }
    }
  };

  v8f acc = {};
  const int nk = (K + 31) / 32;

  stage_A(0, 0);
  stage_W(0, 0);
  asm volatile("s_wait_asynccnt 0" ::: "memory");
  __syncthreads();

  const int idx  = lane & 15;
  const int koff = (lane >> 4) * 2;

  for (int c = 0; c < nk; ++c) {
    const int cur = c & 1;
    if (c + 1 < nk) {          // prefetch next chunk into the other buffer
      stage_A(cur ^ 1, (c + 1) * 32);
      stage_W(cur ^ 1, (c + 1) * 32);
    }
    #pragma unroll
    for (int kk = 0; kk < 32; kk += 4) {
      v2f a, b;
      a.x = As[cur][idx][kk + koff];
      a.y = As[cur][idx][kk + koff + 1];
      b.x = Ws[cur][wave * 16 + idx][kk + koff];
      b.y = Ws[cur][wave * 16 + idx][kk + koff + 1];
      acc = __builtin_amdgcn_wmma_f32_16x16x4_f32(false, a, false, b,
                                                  (short)0, acc, false, false);
    }
    asm volatile("s_wait_asynccnt 0" ::: "memory");
    __syncthreads();
  }

  const int col = n0 + wave * 16 + (lane & 15);
  #pragma unroll
  for (int v = 0; v < 8; ++v) {
    int row = m0 + (lane >> 4) * 8 + v;
    if (row < M && col < N) {
      float r = acc[v];
      if (HAS_BIAS) r += bias[col];
      r = apply_act<ACT>(r);
      if (ACC) C[(long)row * ldc + col] += r;
      else     C[(long)row * ldc + col]  = r;
    }
  }
}

// ---------------------------------------------------------------------------
// adaLN-style modulation: out = sigmoid(g) * (x * rms * (1+s) + sh)
// mod[b] = [s(768) | sh(768) | g(768)] ; one block (256 thr) per row.
// ---------------------------------------------------------------------------
__global__ void __launch_bounds__(256)
ada_kernel(const float* __restrict__ x, const float* __restrict__ mod,
           float* __restrict__ out, int L)
{
  int row = blockIdx.x;
  int b = row / L;
  __shared__ float red[8];
  __shared__ float rmsv;
  float ss = 0.f;
  for (int d = threadIdx.x; d < DM; d += 256) { float v = x[row*DM + d]; ss += v*v; }
  for (int off = 16; off > 0; off >>= 1) ss += __shfl_down(ss, off);
  if ((threadIdx.x & 31) == 0) red[threadIdx.x >> 5] = ss;
  __syncthreads();
  if (threadIdx.x == 0) {
    float s = 0.f; for (int i = 0; i < 8; ++i) s += red[i];
    rmsv = rsqrtf(s / (float)DM + 1e-6f);
  }
  __syncthreads();
  float rms = rmsv;
  for (int d = threadIdx.x; d < DM; d += 256) {
    float s  = mod[b*3*DM + d];
    float sh = mod[b*3*DM + DM + d];
    float g  = mod[b*3*DM + 2*DM + d];
    float v  = x[row*DM + d];
    out[row*DM + d] = sigmoidf_(g) * (v * rms * (1.f + s) + sh);
  }
}

// ---------------------------------------------------------------------------
// LayerNorm; remap=1 gathers the action rows (skip 2 prefix tokens).
// ---------------------------------------------------------------------------
__global__ void __launch_bounds__(256)
ln_kernel(const float* __restrict__ x, const float* __restrict__ g,
          const float* __restrict__ bb, float* __restrict__ out, int remap)
{
  int r = blockIdx.x;
  int src = remap ? ((r / CHUNK) * LTOK + 2 + (r % CHUNK)) : r;
  __shared__ float red[8];
  __shared__ float stats[2];
  float s = 0.f;
  for (int d = threadIdx.x; d < DM; d += 256) s += x[src*DM + d];
  for (int off = 16; off > 0; off >>= 1) s += __shfl_down(s, off);
  if ((threadIdx.x & 31) == 0) red[threadIdx.x >> 5] = s;
  __syncthreads();
  if (threadIdx.x == 0) { float t = 0.f; for (int i = 0; i < 8; ++i) t += red[i]; stats[0] = t / (float)DM; }
  __syncthreads();
  float mu = stats[0];
  float v = 0.f;
  for (int d = threadIdx.x; d < DM; d += 256) { float q = x[src*DM + d] - mu; v += q*q; }
  for (int off = 16; off > 0; off >>= 1) v += __shfl_down(v, off);
  __syncthreads();
  if ((threadIdx.x & 31) == 0) red[threadIdx.x >> 5] = v;
  __syncthreads();
  if (threadIdx.x == 0) { float t = 0.f; for (int i = 0; i < 8; ++i) t += red[i]; stats[1] = rsqrtf(t / (float)DM + 1e-5f); }
  __syncthreads();
  float rs = stats[1];
  for (int d = threadIdx.x; d < DM; d += 256)
    out[r*DM + d] = (x[src*DM + d] - mu) * rs * g[d] + bb[d];
}

// ---------------------------------------------------------------------------
// timestep sinusoid embedding
// ---------------------------------------------------------------------------
__global__ void timestep_kernel(const float* __restrict__ t, float* __restrict__ emb)
{
  int i = blockIdx.x * blockDim.x + threadIdx.x;
  if (i >= Bsz * DM) return;
  int b = i / DM, d = i % DM;
  int h = d % (DM / 2);
  float f = expf(-logf(10000.f) * (float)h / (float)(DM / 2));
  float arg = t[b] * f;
  emb[i] = (d < DM / 2) ? cosf(arg) : sinf(arg);
}

// ---------------------------------------------------------------------------
// assemble x = [proprio, embod, at + pos + t_emb]
// ---------------------------------------------------------------------------
__global__ void build_x_kernel(const float* __restrict__ at, const float* __restrict__ pos,
                               const float* __restrict__ temb, const float* __restrict__ prop,
                               const float* __restrict__ embod, float* __restrict__ x)
{
  int i = blockIdx.x * blockDim.x + threadIdx.x;
  if (i >= Bsz * LTOK * DM) return;
  int d = i % DM, row = i / DM, b = row / LTOK, l = row % LTOK;
  float v;
  if (l == 0)      v = prop[b*DM + d];
  else if (l == 1) v = embod[b*DM + d];
  else { int tt = l - 2; v = at[(b*CHUNK + tt)*DM + d] + pos[tt*DM + d] + temb[b*DM + d]; }
  x[i] = v;
}

// ---------------------------------------------------------------------------
// depthwise causal conv (width 4) + SiLU. xz = in_proj output [rows,3072].
// ---------------------------------------------------------------------------
__global__ void conv_silu_kernel(const float* __restrict__ xz, const float* __restrict__ cw,
                                 const float* __restrict__ cb, float* __restrict__ xmc)
{
  int i = blockIdx.x * blockDim.x + threadIdx.x;
  if (i >= Bsz * LTOK * DIN) return;
  int d = i % DIN, row = i / DIN, l = row % LTOK;
  float s = cb[d];
  #pragma unroll
  for (int k = 0; k < 4; ++k) {
    int ls = l - 3 + k;
    if (ls >= 0) s += xz[(long)(row - l + ls) * (2*DIN) + d] * cw[d*4 + k];
  }
  xmc[i] = s * sigmoidf_(s);
}

// ---------------------------------------------------------------------------
// selective scan: one thread owns (b,d), sequential L=26, n=0..63 outer.
// out = (scan_y + xm*D) * silu(z)
// ---------------------------------------------------------------------------
__global__ void __launch_bounds__(128)
mamba_scan_kernel(const float* __restrict__ xmc, const float* __restrict__ xz,
                  const float* __restrict__ dtb, const float* __restrict__ xp,
                  const float* __restrict__ A_log, const float* __restrict__ Dp,
                  float* __restrict__ out)
{
  int i = blockIdx.x * blockDim.x + threadIdx.x;
  if (i >= Bsz * DIN) return;
  int d = i % DIN, b = i / DIN;
  float dtv[LTOK], xv[LTOK], y[LTOK];
  #pragma unroll
  for (int l = 0; l < LTOK; ++l) {
    int row = b*LTOK + l;
    dtv[l] = dtb[(long)row*DIN + d];
    xv[l]  = xmc[(long)row*DIN + d];
    y[l] = 0.f;
  }
  for (int n = 0; n < DST; ++n) {
    float a = -expf(A_log[(long)d*DST + n]);
    float s = 0.f;
    #pragma unroll
    for (int l = 0; l < LTOK; ++l) {
      int row = b*LTOK + l;
      float Bv = xp[(long)row*XPN + DTR + n];
      float Cv = xp[(long)row*XPN + DTR + DST + n];
      s = expf(a * dtv[l]) * s + dtv[l] * Bv * xv[l];
      y[l] += s * Cv;
    }
  }
  float Dd = Dp[d];
  #pragma unroll
  for (int l = 0; l < LTOK; ++l) {
    int row = b*LTOK + l;
    float z = xz[(long)row*(2*DIN) + DIN + d];
    out[(long)row*DIN + d] = (y[l] + xv[l]*Dd) * (z * sigmoidf_(z));
  }
}

// ---------------------------------------------------------------------------
// attention: scores+softmax (thread per (b,h,query)), then P*V apply.
// ---------------------------------------------------------------------------
template<int LK>
__global__ void attn_scores_kernel(const float* __restrict__ Q, int ldq, int qoff,
                                   const float* __restrict__ Kp, int ldk, int koff,
                                   float* __restrict__ P, int Lq)
{
  int i = blockIdx.x * blockDim.x + threadIdx.x;
  if (i >= Bsz * NH * Lq) return;
  int l = i % Lq, h = (i / Lq) % NH, b = i / (Lq * NH);
  const float* q = Q + (long)(b*Lq + l)*ldq + qoff + h*HD;
  float sc[LK];
  float mx = -1e30f;
  #pragma unroll 1
  for (int j = 0; j < LK; ++j) {
    const float* k = Kp + (long)(b*LK + j)*ldk + koff + h*HD;
    float s = 0.f;
    for (int d2 = 0; d2 < HD; ++d2) s += q[d2]*k[d2];
    s *= 0.125f;
    sc[j] = s; mx = fmaxf(mx, s);
  }
  float sum = 0.f;
  #pragma unroll
  for (int j = 0; j < LK; ++j) { sc[j] = expf(sc[j] - mx); sum += sc[j]; }
  float inv = 1.f / sum;
  #pragma unroll
  for (int j = 0; j < LK; ++j) P[(long)i*LK + j] = sc[j] * inv;
}

template<int LK>
__global__ void attn_apply_kernel(const float* __restrict__ P, const float* __restrict__ V,
                                  int ldv, int voff, float* __restrict__ Out, int Lq)
{
  int i = blockIdx.x * blockDim.x + threadIdx.x;
  if (i >= Bsz * NH * Lq * HD) return;
  int d = i % HD, l = (i / HD) % Lq, h = (i / (HD*Lq)) % NH, b = i / (HD*Lq*NH);
  const float* p = P + (long)((b*NH + h)*Lq + l) * LK;
  float s = 0.f;
  #pragma unroll
  for (int j = 0; j < LK; ++j) s += p[j] * V[(long)(b*LK + j)*ldv + voff + h*HD + d];
  Out[(long)(b*Lq + l)*DM + h*HD + d] = s;
}

// ---------------------------------------------------------------------------
// host orchestration
// ---------------------------------------------------------------------------
static void gemm(hipStream_t st, const float* A, int lda, const float* W,
                 const float* bias, float* C, int ldc, int M, int N, int K,
                 int act, bool accf)
{
  dim3 g((N + 127) / 128, (M + 15) / 16), b(256);
  if (accf) {
    if (bias) gemm_wmma<0, true,  true ><<<g, b, 0, st>>>(A, lda, W, bias, C, ldc, M, N, K);
    else      gemm_wmma<0, true,  false><<<g, b, 0, st>>>(A, lda, W, bias, C, ldc, M, N, K);
  } else if (act == 1) {
    gemm_wmma<1, false, true ><<<g, b, 0, st>>>(A, lda, W, bias, C, ldc, M, N, K);
  } else if (act == 2) {
    gemm_wmma<2, false, true ><<<g, b, 0, st>>>(A, lda, W, bias, C, ldc, M, N, K);
  } else if (act == 3) {
    gemm_wmma<3, false, true ><<<g, b, 0, st>>>(A, lda, W, bias, C, ldc, M, N, K);
  } else {
    if (bias) gemm_wmma<0, false, true ><<<g, b, 0, st>>>(A, lda, W, bias, C, ldc, M, N, K);
    else      gemm_wmma<0, false, false><<<g, b, 0, st>>>(A, lda, W, bias, C, ldc, M, N, K);
  }
}

static inline int ew(int n) { return (n + 255) / 256; }

extern "C" void kernel_launch(void* const* d_in, const int* in_sizes, int n_in,
                              void* d_out, int out_size, void* d_ws, size_t ws_size,
                              hipStream_t stream)
{
  (void)in_sizes; (void)n_in; (void)out_size; (void)ws_size;
  auto F = [&](int i) { return (const float*)d_in[i]; };

  // ---- tensor inputs (setup_inputs order) ----
  const float* noisy = F(0);      // [4,24,14]
  const float* flowt = F(1);      // [4]
  const float* condp = F(2);      // [4,32,1024]
  const float* prop  = F(3);      // [4,768]
  const float* embod = F(4);      // [4,768]

  // ---- params: pytree flattened, dict keys sorted, {'b','w'} -> b,w ----
  const int P0 = 5;
  const int ACT_B = P0 + 0, ACT_W = P0 + 1;
  auto ATT = [&](int i, int leaf) { return P0 + 2 + 24*i + leaf; };
  // attn leaves: 0 ck.b 1 ck.w 2 co.b 3 co.w 4 cq.b 5 cq.w 6 cv.b 7 cv.w
  //              8 ffn1.b 9 ffn1.w 10 ffn2.b 11 ffn2.w 12 nckv.b 13 nckv.g
  //              14 ncq.b 15 ncq.w 16 nffn.b 17 nffn.w 18 nself.b 19 nself.w
  //              20 qkv.b 21 qkv.w 22 so.b 23 so.w
  const int CONDP_B = P0 + 2 + 144, CONDP_W = CONDP_B + 1;
  const int EMB_B = CONDP_W + 1, EMB_W = EMB_B + 1;
  auto MAM = [&](int j, int leaf) { return EMB_W + 1 + 11*j + leaf; };
  // mamba leaves: 0 A_log 1 D 2 conv_b 3 conv_w 4 dt.b 5 dt.w 6 in_proj.w
  //               7 norm.b 8 norm.w 9 out_proj.w 10 x_proj.w
  const int AFTER_MAM = EMB_W + 1 + 11*12;
  const int ONORM_B = AFTER_MAM + 0, ONORM_G = AFTER_MAM + 1;
  const int OPROJ_B = AFTER_MAM + 2, OPROJ_W = AFTER_MAM + 3;
  const int POS_EMB = AFTER_MAM + 4;
  const int PROP_B = AFTER_MAM + 5, PROP_W = AFTER_MAM + 6;
  const int TC1_B = AFTER_MAM + 7, TC1_W = AFTER_MAM + 8;
  const int TC2_B = AFTER_MAM + 9, TC2_W = AFTER_MAM + 10;
  const int TE1_B = AFTER_MAM + 11, TE1_W = AFTER_MAM + 12;
  const int TE2_B = AFTER_MAM + 13, TE2_W = AFTER_MAM + 14;

  // ---- workspace layout (fp32) ----
  float* ws = (float*)d_ws;
  size_t off = 0;
  auto alloc = [&](size_t n) { float* p = ws + off; off += n; return p; };
  const int ROWS = Bsz * LTOK;            // 104
  float* xbuf   = alloc((size_t)ROWS * DM);
  float* sinemb = alloc(Bsz * DM);
  float* mlp4h  = alloc(Bsz * MLPH);
  float* temb   = alloc(Bsz * DM);
  float* tch    = alloc(Bsz * DM);
  float* tcond  = alloc(Bsz * DM);
  float* prop_o = alloc(Bsz * DM);
  float* emb_o  = alloc(Bsz * DM);
  float* atb    = alloc((size_t)Bsz * CHUNK * DM);
  float* cond   = alloc((size_t)Bsz * 32 * DM);
  float* kvin   = alloc((size_t)Bsz * 32 * DM);
  float* modb   = alloc(Bsz * 3 * DM);
  float* h1     = alloc((size_t)ROWS * DM);
  float* big    = alloc((size_t)ROWS * MLPH);  // in_proj out / qkv / ffn hidden
  float* xmc    = alloc((size_t)ROWS * DIN);
  float* xp     = alloc((size_t)ROWS * XPN);
  float* dtb    = alloc((size_t)ROWS * DIN);
  float* ybuf   = alloc((size_t)ROWS * DIN);
  float* qb     = alloc((size_t)ROWS * DM);
  float* kb     = alloc((size_t)Bsz * 32 * DM);
  float* vb     = alloc((size_t)Bsz * 32 * DM);
  float* cob    = alloc((size_t)ROWS * DM);
  float* pb     = alloc((size_t)Bsz * NH * LTOK * 32);
  float* xa     = alloc((size_t)Bsz * CHUNK * DM);

  // ---- embeddings / conditioning ----
  timestep_kernel<<<ew(Bsz*DM), 256, 0, stream>>>(flowt, sinemb);
  gemm(stream, sinemb, DM, F(TE1_W), F(TE1_B), mlp4h, MLPH, Bsz, MLPH, DM, 1, false);
  gemm(stream, mlp4h, MLPH, F(TE2_W), F(TE2_B), temb, DM, Bsz, DM, MLPH, 0, false);
  gemm(stream, temb, DM, F(TC1_W), F(TC1_B), tch, DM, Bsz, DM, DM, 1, false);
  gemm(stream, tch, DM, F(TC2_W), F(TC2_B), tcond, DM, Bsz, DM, DM, 0, false);
  gemm(stream, noisy, 14, F(ACT_W), F(ACT_B), atb, DM, Bsz*CHUNK, DM, 14, 0, false);
  gemm(stream, prop, DM, F(PROP_W), F(PROP_B), prop_o, DM, Bsz, DM, DM, 0, false);
  gemm(stream, embod, DM, F(EMB_W), F(EMB_B), emb_o, DM, Bsz, DM, DM, 0, false);
  build_x_kernel<<<ew(ROWS*DM), 256, 0, stream>>>(atb, F(POS_EMB), temb, prop_o, emb_o, xbuf);
  gemm(stream, condp, 1024, F(CONDP_W), F(CONDP_B), cond, DM, Bsz*32, DM, 1024, 0, false);

  // ---- 18-layer pattern: (mamba, mamba, attn) x 6 ----
  int mi = 0, ai = 0;
  for (int layer = 0; layer < 18; ++layer) {
    if (layer % 3 != 2) {
      int j = mi++;
      gemm(stream, tcond, DM, F(MAM(j,8)), F(MAM(j,7)), modb, 3*DM, Bsz, 3*DM, DM, 0, false);
      ada_kernel<<<ROWS, 256, 0, stream>>>(xbuf, modb, h1, LTOK);
      gemm(stream, h1, DM, F(MAM(j,6)), nullptr, big, 2*DIN, ROWS, 2*DIN, DM, 0, false);
      conv_silu_kernel<<<ew(ROWS*DIN), 256, 0, stream>>>(big, F(MAM(j,3)), F(MAM(j,2)), xmc);
      gemm(stream, xmc, DIN, F(MAM(j,10)), nullptr, xp, XPN, ROWS, XPN, DIN, 0, false);
      gemm(stream, xp, XPN, F(MAM(j,5)), F(MAM(j,4)), dtb, DIN, ROWS, DIN, DTR, 3, false);
      mamba_scan_kernel<<<(Bsz*DIN + 127)/128, 128, 0, stream>>>(xmc, big, dtb, xp,
                                                                 F(MAM(j,0)), F(MAM(j,1)), ybuf);
      gemm(stream, ybuf, DIN, F(MAM(j,9)), nullptr, xbuf, DM, ROWS, DM, DIN, 0, true);
    } else {
      int i = ai++;
      // cross-attention
      gemm(stream, tcond, DM, F(ATT(i,15)), F(ATT(i,14)), modb, 3*DM, Bsz, 3*DM, DM, 0, false);
      ada_kernel<<<ROWS, 256, 0, stream>>>(xbuf, modb, h1, LTOK);
      gemm(stream, h1, DM, F(ATT(i,5)), F(ATT(i,4)), qb, DM, ROWS, DM, DM, 0, false);
      ln_kernel<<<Bsz*32, 256, 0, stream>>>(cond, F(ATT(i,13)), F(ATT(i,12)), kvin, 0);
      gemm(stream, kvin, DM, F(ATT(i,1)), F(ATT(i,0)), kb, DM, Bsz*32, DM, DM, 0, false);
      gemm(stream, kvin, DM, F(ATT(i,7)), F(ATT(i,6)), vb, DM, Bsz*32, DM, DM, 0, false);
      attn_scores_kernel<32><<<ew(Bsz*NH*LTOK), 256, 0, stream>>>(qb, DM, 0, kb, DM, 0, pb, LTOK);
      attn_apply_kernel<32><<<ew(Bsz*NH*LTOK*HD), 256, 0, stream>>>(pb, vb, DM, 0, cob, LTOK);
      gemm(stream, cob, DM, F(ATT(i,3)), F(ATT(i,2)), xbuf, DM, ROWS, DM, DM, 0, true);
      // self-attention
      gemm(stream, tcond, DM, F(ATT(i,19)), F(ATT(i,18)), modb, 3*DM, Bsz, 3*DM, DM, 0, false);
      ada_kernel<<<ROWS, 256, 0, stream>>>(xbuf, modb, h1, LTOK);
      gemm(stream, h1, DM, F(ATT(i,21)), F(ATT(i,20)), big, 3*DM, ROWS, 3*DM, DM, 0, false);
      attn_scores_kernel<LTOK><<<ew(Bsz*NH*LTOK), 256, 0, stream>>>(big, 3*DM, 0, big, 3*DM, DM, pb, LTOK);
      attn_apply_kernel<LTOK><<<ew(Bsz*NH*LTOK*HD), 256, 0, stream>>>(pb, big, 3*DM, 2*DM, cob, LTOK);
      gemm(stream, cob, DM, F(ATT(i,23)), F(ATT(i,22)), xbuf, DM, ROWS, DM, DM, 0, true);
      // FFN
      gemm(stream, tcond, DM, F(ATT(i,17)), F(ATT(i,16)), modb, 3*DM, Bsz, 3*DM, DM, 0, false);
      ada_kernel<<<ROWS, 256, 0, stream>>>(xbuf, modb, h1, LTOK);
      gemm(stream, h1, DM, F(ATT(i,9)), F(ATT(i,8)), big, MLPH, ROWS, MLPH, DM, 2, false);
      gemm(stream, big, MLPH, F(ATT(i,11)), F(ATT(i,10)), xbuf, DM, ROWS, DM, MLPH, 0, true);
    }
  }

  // ---- output head: LN over action rows, project to 14 dims ----
  ln_kernel<<<Bsz*CHUNK, 256, 0, stream>>>(xbuf, F(ONORM_G), F(ONORM_B), xa, 1);
  gemm(stream, xa, DM, F(OPROJ_W), F(OPROJ_B), (float*)d_out, 14, Bsz*CHUNK, 14, DM, 0, false);
}